// FRP_Module_61014305407289
// MI455X (gfx1250) — compile-verified
//
#include <hip/hip_runtime.h>
#include <hip/hip_bf16.h>

// ---------------------------------------------------------------------------
// FRP module for MI455X (gfx1250).
// Shapes fixed by the reference: B=8, C=3, T=64, H=W=224, W_WIN=5.
//
// HBM-bound: >=725MB traffic @23.3TB/s => ~31us floor. Two passes over x
// (recompute in pass2 instead of materializing `re`, saving 308MB of traffic).
// The banded temporal einsum is executed on the matrix pipe with
// V_WMMA_F32_16X16X4_F32; A (Wmat tiles) is synthesized in registers.
// ---------------------------------------------------------------------------

#define T_FRAMES 64
#define HW       50176          // 224*224
#define CCH      3
#define BATCH    8
#define W_WIN    5
#define T_OUT    (T_FRAMES + W_WIN - 1)   // 68

typedef float v2f __attribute__((ext_vector_type(2)));
typedef float v4f __attribute__((ext_vector_type(4)));
typedef float v8f __attribute__((ext_vector_type(8)));

__device__ __forceinline__ float gelu_exact(float x) {
    // torch F.gelu (exact / erf form)
    return 0.5f * x * (1.0f + erff(x * 0.70710678118654752440f));
}

// Monotonic float<->uint32 encoding so min/max reductions can use integer
// atomics (order-independent => deterministic across graph replays).
__device__ __forceinline__ unsigned enc_ord(float f) {
    unsigned u = __float_as_uint(f);
    return (u & 0x80000000u) ? ~u : (u | 0x80000000u);
}
__device__ __forceinline__ float dec_ord(unsigned e) {
    unsigned u = (e & 0x80000000u) ? (e & 0x7FFFFFFFu) : ~e;
    return __uint_as_float(u);
}

// Compute the three per-channel 16x16 (t x hw) tiles D[c] = Wmat_tile @ X_tile
// for t-tile `mt`, column base hw0, using chained f32 WMMAs over the band.
//   A layout (16x4, MxK): lane M=lane&15; VGPR0 holds K=2*half, VGPR1 K=2*half+1
//   B layout (4x16, KxN): lane N=lane&15; VGPR0 holds K=2*half, VGPR1 K=2*half+1
//   D layout (16x16):     lane N=lane&15; VGPR v holds M = v + 8*half
__device__ __forceinline__ void compute_tiles(const float* __restrict__ xg,
                                              int b, int hw0, int mt, int lane,
                                              v8f* D) {
    const int hh = lane >> 4;        // half-wave
    const int n  = lane & 15;
    const int t_row = 16 * mt + n;   // A-matrix row for this lane
    const int Ncl = min(W_WIN, T_FRAMES - t_row);   // truncated window length

    const v8f zero = {0.f, 0.f, 0.f, 0.f, 0.f, 0.f, 0.f, 0.f};
    D[0] = zero; D[1] = zero; D[2] = zero;

    // Band coverage: rows 16mt..16mt+15 only touch s in [16mt, 16mt+19]
    // => K-chunks 4mt..4mt+4 (mt=3: truncated windows stop at s=63 => 4 chunks)
    const int kc_lo = 4 * mt;
    const int kc_hi = (mt == 3) ? 15 : (4 * mt + 4);

    for (int kc = kc_lo; kc <= kc_hi; ++kc) {
        const int s0 = 4 * kc + 2 * hh;           // K handled by this lane half
        // Rank-pooling weight: Wmat[t, s] = 2*(j+1)-N-1 for j = s-t in [0,N)
        const int j0 = s0 - t_row;
        const int j1 = j0 + 1;
        v2f A;
        A.x = (j0 >= 0 && j0 < Ncl) ? (float)(2 * (j0 + 1) - Ncl - 1) : 0.0f;
        A.y = (j1 >= 0 && j1 < Ncl) ? (float)(2 * (j1 + 1) - Ncl - 1) : 0.0f;

        const size_t base =
            ((size_t)b * CCH * T_FRAMES + (size_t)s0) * HW + (size_t)hw0 + n;

        if (kc < kc_hi) {   // pull next chunk's lines toward L2 (global_prefetch_b8)
            __builtin_prefetch(xg + base + 4 * (size_t)HW, 0, 0);
        }

#pragma unroll
        for (int c = 0; c < CCH; ++c) {
            const size_t bc = base + (size_t)c * T_FRAMES * HW;
            v2f Bv;
            Bv.x = xg[bc];          // frame s0,   16 contiguous floats/half-wave
            Bv.y = xg[bc + HW];     // frame s0+1
            D[c] = __builtin_amdgcn_wmma_f32_16x16x4_f32(
                false, A, false, Bv, (short)0, D[c], false, false);
        }
    }
}

// ---------------------------------------------------------------------------
// Kernel 0: zero the 4-frame mask region of the output and init the encoded
// min/max accumulators in workspace.
// ---------------------------------------------------------------------------
__global__ void k_init(float* __restrict__ out,
                       unsigned* __restrict__ gmin,
                       unsigned* __restrict__ gmax) {
    const size_t maskHW = (size_t)(W_WIN - 1) * HW;        // 200704 floats / batch
    const size_t total4 = (size_t)BATCH * maskHW / 4;      // v4f stores
    size_t idx = (size_t)blockIdx.x * blockDim.x + threadIdx.x;
    if (idx < total4) {
        size_t e = idx * 4;
        size_t b = e / maskHW;
        size_t r = e % maskHW;
        v4f z = {0.f, 0.f, 0.f, 0.f};
        __builtin_nontemporal_store(z, (v4f*)(out + b * (size_t)T_OUT * HW + r));
    }
    if (blockIdx.x == 0 && threadIdx.x < T_FRAMES) {
        gmin[threadIdx.x] = 0xFFFFFFFFu;   // enc(+inf-ish): identity for min
        gmax[threadIdx.x] = 0u;            // identity for max
    }
}

// ---------------------------------------------------------------------------
// Pass 1: WMMA window-sum + GELU, reduce per-t min/max (wave shfl -> LDS
// integer atomics -> 128 global atomics per block).
// Block = 256 thr = 8 waves; each wave owns 16 hw columns. Grid (392, 8).
// ---------------------------------------------------------------------------
__global__ void __launch_bounds__(256)
k_pass1(const float* __restrict__ xg,
        unsigned* __restrict__ gmin, unsigned* __restrict__ gmax) {
    __shared__ unsigned smin[T_FRAMES];
    __shared__ unsigned smax[T_FRAMES];
    const int tid = threadIdx.x;
    if (tid < T_FRAMES) { smin[tid] = 0xFFFFFFFFu; smax[tid] = 0u; }
    __syncthreads();

    const int wave = tid >> 5;
    const int lane = tid & 31;
    const int hh   = lane >> 4;
    const int n    = lane & 15;
    const int b    = blockIdx.y;
    const int hw0  = (blockIdx.x * 8 + wave) * 16;

    for (int mt = 0; mt < 4; ++mt) {
        v8f D[CCH];
        compute_tiles(xg, b, hw0, mt, lane, D);
#pragma unroll
        for (int v = 0; v < 8; ++v) {
            float g0 = gelu_exact(D[0][v]);
            float g1 = gelu_exact(D[1][v]);
            float g2 = gelu_exact(D[2][v]);
            float mnv = fminf(g0, fminf(g1, g2));
            float mxv = fmaxf(g0, fmaxf(g1, g2));
            // butterfly across the 16 columns (stays within each half-wave)
#pragma unroll
            for (int msk = 8; msk >= 1; msk >>= 1) {
                mnv = fminf(mnv, __shfl_xor(mnv, msk, 32));
                mxv = fmaxf(mxv, __shfl_xor(mxv, msk, 32));
            }
            if (n == 0) {
                int t = 16 * mt + v + 8 * hh;   // D-tile row for this VGPR/half
                atomicMin(&smin[t], enc_ord(mnv));
                atomicMax(&smax[t], enc_ord(mxv));
            }
        }
    }
    __syncthreads();
    if (tid < T_FRAMES) {
        atomicMin(&gmin[tid], smin[tid]);
        atomicMax(&gmax[tid], smax[tid]);
    }
}

// ---------------------------------------------------------------------------
// Finalize: per-t (mn, 1/denom), matching the reference's
// denom = (max-min)!=0 ? (max-min) : 1e-5 rule.
// ---------------------------------------------------------------------------
__global__ void k_finalize(const unsigned* __restrict__ gmin,
                           const unsigned* __restrict__ gmax,
                           float* __restrict__ mninv) {
    int t = threadIdx.x;
    if (t < T_FRAMES) {
        float mn = dec_ord(gmin[t]);
        float mx = dec_ord(gmax[t]);
        float remx = mx - mn;
        float denom = (remx != 0.0f) ? remx : 1e-5f;
        mninv[t] = mn;
        mninv[T_FRAMES + t] = 1.0f / denom;
    }
}

// ---------------------------------------------------------------------------
// Pass 2: recompute WMMA window-sum + GELU, normalize per-t, grayscale dot,
// stream the result with non-temporal stores into out[b, 0, 4+t, h, w].
// sum_c gw[c]*(g_c - mn)/denom == ( sum_c gw[c]*g_c - mn*sum(gw) ) * inv.
// ---------------------------------------------------------------------------
__global__ void __launch_bounds__(256)
k_pass2(const float* __restrict__ xg,
        const float* __restrict__ mninv,
        float* __restrict__ out) {
    __shared__ float s_mn[T_FRAMES];
    __shared__ float s_inv[T_FRAMES];
    const int tid = threadIdx.x;
    if (tid < 2 * T_FRAMES) {
        float vv = mninv[tid];
        if (tid < T_FRAMES) s_mn[tid] = vv;
        else                s_inv[tid - T_FRAMES] = vv;
    }
    __syncthreads();

    const int wave = tid >> 5;
    const int lane = tid & 31;
    const int hh   = lane >> 4;
    const int n    = lane & 15;
    const int b    = blockIdx.y;
    const int hw0  = (blockIdx.x * 8 + wave) * 16;

    const float GW0 = 0.2989f, GW1 = 0.587f, GW2 = 0.114f;
    const float GWS = GW0 + GW1 + GW2;

    for (int mt = 0; mt < 4; ++mt) {
        v8f D[CCH];
        compute_tiles(xg, b, hw0, mt, lane, D);
#pragma unroll
        for (int v = 0; v < 8; ++v) {
            float dot = GW0 * gelu_exact(D[0][v])
                      + GW1 * gelu_exact(D[1][v])
                      + GW2 * gelu_exact(D[2][v]);
            int t = 16 * mt + v + 8 * hh;
            float val = (dot - s_mn[t] * GWS) * s_inv[t];
            size_t oidx = ((size_t)b * T_OUT + (W_WIN - 1) + t) * HW
                        + (size_t)hw0 + n;
            __builtin_nontemporal_store(val, out + oidx);
        }
    }
}

// ---------------------------------------------------------------------------
extern "C" void kernel_launch(void* const* d_in, const int* in_sizes, int n_in,
                              void* d_out, int out_size, void* d_ws, size_t ws_size,
                              hipStream_t stream) {
    (void)in_sizes; (void)n_in; (void)out_size; (void)ws_size;
    const float* x = (const float*)d_in[0];
    float* out = (float*)d_out;

    // workspace layout: [64]u32 enc-min | [64]u32 enc-max | [128]f32 (mn, inv)
    unsigned* gmin  = (unsigned*)d_ws;
    unsigned* gmax  = gmin + T_FRAMES;
    float*    mninv = (float*)(gmax + T_FRAMES);

    const int initTotal4 = (BATCH * (W_WIN - 1) * HW) / 4;   // 401408
    const int initBlocks = (initTotal4 + 255) / 256;         // 1568
    k_init<<<initBlocks, 256, 0, stream>>>(out, gmin, gmax);

    dim3 grid(HW / (8 * 16), BATCH);   // (392, 8); 8 waves x 16 cols per block
    k_pass1<<<grid, 256, 0, stream>>>(x, gmin, gmax);
    k_finalize<<<1, T_FRAMES, 0, stream>>>(gmin, gmax, mninv);
    k_pass2<<<grid, 256, 0, stream>>>(x, mninv, out);
}